// Attention_15229954031958
// MI455X (gfx1250) — compile-verified
//
#include <hip/hip_runtime.h>
#include <cstdint>

// Paged attention decode (GQA) for gfx1250.
// B=32, H=32, D=128, KVH=8, G=4, PAGE=16, MAX_BLOCKS=256.
// One workgroup per (batch, kv_head); 8 waves; each wave flash-decodes a
// strided subset of pages with V_WMMA_F32_16X16X4_F32, K/V staged into LDS
// via global_load_async_to_lds_b128; cross-wave LSE combine at the end.

#define B_     32
#define H_     32
#define D_     128
#define KVH_   8
#define G_     4
#define PAGE_  16
#define MB_    256
#define NW_    8          // waves per workgroup
#define KP_    132        // K/V tile row pitch (floats), pad for banks
#define PP_    17         // P tile row pitch (floats)
#define SCALE_ 0.08838834764831845f
#define NEG_   (-1.0e30f)

typedef float v2f __attribute__((ext_vector_type(2)));
typedef float v8f __attribute__((ext_vector_type(8)));

// floats per wave region: K tile + V tile + P tile + alpha scratch
#define WREG_ (2 * 16 * KP_ + 16 * PP_ + 16)
#define SMEM_FLOATS_ (16 * KP_ + NW_ * WREG_)
#define SMEM_BYTES_ (SMEM_FLOATS_ * 4)

__global__ __launch_bounds__(256) void
pa_decode_kernel(const float* __restrict__ q,
                 const float* __restrict__ kc,
                 const float* __restrict__ vc,
                 const int* __restrict__ bt,
                 const int* __restrict__ cl,
                 float* __restrict__ out) {
  extern __shared__ float smem[];

  const int blk = blockIdx.x;
  const int b = blk >> 3;          // batch
  const int kvh = blk & 7;         // kv head
  const int tid = threadIdx.x;
  const int w = tid >> 5;          // wave id 0..7
  const int lane = tid & 31;
  const int n = lane & 15;         // fragment column / M-row index
  const int hh = lane >> 4;        // lane half (K-pair select)

  float* Qs = smem;                                  // [16][KP_] (rows>=4 zero)
  float* wbase = smem + 16 * KP_ + w * WREG_;
  float* Kt = wbase;                                 // [16][KP_]
  float* Vt = wbase + 16 * KP_;                      // [16][KP_]
  float* Pt = wbase + 2 * 16 * KP_;                  // [16][PP_]
  float* Al = Pt + 16 * PP_;                         // [4] alpha per head

  // ---- stage Q (pre-scaled by 1/sqrt(D)), zero-pad to 16 rows ----
  for (int i = tid; i < 16 * KP_; i += 256) {
    int r = i / KP_;
    int c = i - r * KP_;
    float v = 0.0f;
    if (r < G_ && c < D_) v = q[(b * H_ + kvh * G_ + r) * D_ + c] * SCALE_;
    Qs[i] = v;
  }
  __syncthreads();

  const int ctx = cl[b];
  const int npages = (ctx + PAGE_ - 1) / PAGE_;

  // per-lane softmax state layout: head group hd = lane>>3 owns tokens tk,tk+8
  const int hd = lane >> 3;
  const int tk = lane & 7;
  float m_run = NEG_;
  float l_run = 0.0f;
  v8f co[8] = {};  // O^T accumulators: tile t covers d = 16t..16t+15

  // async copy lane mapping: 2 lanes per token row, 16B chunks, stride 32B
  const int rrow = lane >> 1;
  const int c0 = lane & 1;
  const unsigned ldsK0 = (unsigned)(uintptr_t)(Kt + rrow * KP_ + c0 * 4);
  const unsigned ldsV0 = (unsigned)(uintptr_t)(Vt + rrow * KP_ + c0 * 4);

  for (int p = w; p < npages; p += NW_) {
    const int phys = bt[b * MB_ + p];
    // global byte addresses for this lane's first 16B chunk of K / V rows
    unsigned long long gk =
        (unsigned long long)(uintptr_t)(kc + ((size_t)phys * (PAGE_ * KVH_ * D_) +
                                              (size_t)rrow * (KVH_ * D_) +
                                              (size_t)kvh * D_ + (size_t)c0 * 4));
    unsigned long long gv =
        (unsigned long long)(uintptr_t)(vc + ((size_t)phys * (PAGE_ * KVH_ * D_) +
                                              (size_t)rrow * (KVH_ * D_) +
                                              (size_t)kvh * D_ + (size_t)c0 * 4));
    unsigned lk = ldsK0;
    unsigned lv = ldsV0;
#pragma unroll
    for (int i = 0; i < 16; ++i) {
      asm volatile("global_load_async_to_lds_b128 %0, %1, off"
                   :: "v"(lk), "v"(gk) : "memory");
      asm volatile("global_load_async_to_lds_b128 %0, %1, off"
                   :: "v"(lv), "v"(gv) : "memory");
      lk += 32; lv += 32; gk += 32; gv += 32;
    }
    asm volatile("s_wait_asynccnt 0" ::: "memory");
    __builtin_amdgcn_wave_barrier();

    // ---- S^T = K_tile x Q^T : M=token, N=head, 32 x (16x16x4 f32) ----
    v8f cs = {};
#pragma unroll
    for (int kk = 0; kk < 32; ++kk) {
      v2f a = *(const v2f*)(Kt + n * KP_ + 4 * kk + 2 * hh);
      v2f bq = *(const v2f*)(Qs + n * KP_ + 4 * kk + 2 * hh);
      cs = __builtin_amdgcn_wmma_f32_16x16x4_f32(false, a, false, bq,
                                                 (short)0, cs, false, false);
    }

    // ---- mask + spill scores to LDS (P^T raw) ----
    const int s0 = p * PAGE_;
#pragma unroll
    for (int r = 0; r < 8; ++r) {
      int tok = r + 8 * hh;
      float sc = cs[r];
      if (s0 + tok >= ctx) sc = NEG_;
      Pt[tok * PP_ + n] = sc;
    }
    __builtin_amdgcn_wave_barrier();

    // ---- online softmax per head (group-of-8 lanes per head) ----
    float s_a = Pt[tk * PP_ + hd];
    float s_b = Pt[(tk + 8) * PP_ + hd];
    float vmax = fmaxf(s_a, s_b);
#pragma unroll
    for (int off = 1; off < 8; off <<= 1)
      vmax = fmaxf(vmax, __shfl_xor(vmax, off, 32));
    float m_new = fmaxf(m_run, vmax);
    float alpha = __expf(m_run - m_new);
    float e_a = __expf(s_a - m_new);
    float e_b = __expf(s_b - m_new);
    Pt[tk * PP_ + hd] = e_a;
    Pt[(tk + 8) * PP_ + hd] = e_b;
    float ls = e_a + e_b;
#pragma unroll
    for (int off = 1; off < 8; off <<= 1)
      ls += __shfl_xor(ls, off, 32);
    l_run = l_run * alpha + ls;
    m_run = m_new;
    if (tk == 0) Al[hd] = alpha;
    __builtin_amdgcn_wave_barrier();

    // rescale output accumulators (per-column alpha; cols >= G stay 0)
    float alphaN = (n < G_) ? Al[n] : 0.0f;
#pragma unroll
    for (int t = 0; t < 8; ++t) co[t] = co[t] * alphaN;

    // ---- O^T += V^T x P^T : M = d-tile, K = tokens, N = head ----
#pragma unroll
    for (int t = 0; t < 8; ++t) {
      const int dcol = 16 * t + n;
#pragma unroll
      for (int ks = 0; ks < 4; ++ks) {
        const int r0 = 4 * ks + 2 * hh;
        v2f a;
        a.x = Vt[r0 * KP_ + dcol];
        a.y = Vt[(r0 + 1) * KP_ + dcol];
        v2f bp;
        bp.x = Pt[r0 * PP_ + n];
        bp.y = Pt[(r0 + 1) * PP_ + n];
        co[t] = __builtin_amdgcn_wmma_f32_16x16x4_f32(false, a, false, bp,
                                                      (short)0, co[t], false, false);
      }
    }
  }

  // ---- cross-wave combine (alias dead KV-tile LDS) ----
  __syncthreads();
  float* Oacc = smem + 16 * KP_;               // [NW_][G_][D_]
  float* Mw = Oacc + NW_ * G_ * D_;            // [NW_][G_]
  float* Lw = Mw + NW_ * G_;                   // [NW_][G_]

  if (n < G_) {
#pragma unroll
    for (int t = 0; t < 8; ++t)
#pragma unroll
      for (int j = 0; j < 8; ++j)
        Oacc[(w * G_ + n) * D_ + 16 * t + j + 8 * hh] = co[t][j];
  }
  if (tk == 0) {
    Mw[w * G_ + hd] = m_run;
    Lw[w * G_ + hd] = l_run;
  }
  __syncthreads();

  for (int e = tid; e < G_ * D_; e += 256) {
    const int g = e >> 7;
    const int d = e & (D_ - 1);
    float M = NEG_;
#pragma unroll
    for (int ww = 0; ww < NW_; ++ww) M = fmaxf(M, Mw[ww * G_ + g]);
    float Ls = 0.0f, val = 0.0f;
#pragma unroll
    for (int ww = 0; ww < NW_; ++ww) {
      const float f = __expf(Mw[ww * G_ + g] - M);
      Ls += Lw[ww * G_ + g] * f;
      val += Oacc[(ww * G_ + g) * D_ + d] * f;
    }
    out[(b * H_ + kvh * G_ + g) * D_ + d] = val / Ls;
  }
}

extern "C" void kernel_launch(void* const* d_in, const int* in_sizes, int n_in,
                              void* d_out, int out_size, void* d_ws, size_t ws_size,
                              hipStream_t stream) {
  (void)in_sizes; (void)n_in; (void)d_ws; (void)ws_size; (void)out_size;
  const float* q = (const float*)d_in[0];
  const float* kc = (const float*)d_in[1];
  const float* vc = (const float*)d_in[2];
  const int* bt = (const int*)d_in[3];
  const int* cl = (const int*)d_in[4];
  float* out = (float*)d_out;

  hipFuncSetAttribute((const void*)pa_decode_kernel,
                      hipFuncAttributeMaxDynamicSharedMemorySize, SMEM_BYTES_);
  pa_decode_kernel<<<B_ * KVH_, 256, SMEM_BYTES_, stream>>>(q, kc, vc, bt, cl, out);
}